// HammingL2_21500606284003
// MI455X (gfx1250) — compile-verified
//
#include <hip/hip_runtime.h>

// CDNA5 / gfx1250 — HammingL2 via symmetric Gram-matrix SYRK with
// V_WMMA_F32_16X16X4_F32 and async global->LDS staging.
//
// out = mean_n( sum_i d_i v_{n,i}^2 - v_n^T W v_n ) = (1/N) sum_{ij} E_ij G_ij
// with E = diag(d) - W (symmetric), G = luts^T luts (SYRK, upper triangle x2).

typedef __attribute__((ext_vector_type(2))) float v2f;
typedef __attribute__((ext_vector_type(8))) float v8f;

#define L         256
#define NUM_LUTS  65536
#define KB        64            // rows of luts staged per block (K-chunk)
#define PITCH     (KB + 1)      // odd pitch -> conflict-free transposed LDS
#define NBLK      (NUM_LUTS / KB)

#if __has_builtin(__builtin_amdgcn_global_load_async_to_lds_b32)
#define HL2_ASYNC_STAGE 1
#endif

// Address-space-typed pointers for the async builtin:
//   param 0: addrspace(1) int*  (prints as "__device__ int *")
//   param 1: addrspace(3) int*
typedef __attribute__((address_space(1))) float* gasf_t;
typedef __attribute__((address_space(1))) int*   gasi_t;
typedef __attribute__((address_space(3))) float* lasf_t;
typedef __attribute__((address_space(3))) int*   lasi_t;

__device__ __forceinline__ void hl2_sched_fence() {
#if __has_builtin(__builtin_amdgcn_sched_barrier)
  __builtin_amdgcn_sched_barrier(0);   // keep tile-group live ranges separate
#endif
}

__device__ __forceinline__ void hl2_wait_async() {
#ifdef HL2_ASYNC_STAGE
#if __has_builtin(__builtin_amdgcn_s_wait_asynccnt)
  __builtin_amdgcn_s_wait_asynccnt(0);
#else
  asm volatile("s_wait_asynccnt 0x0" ::: "memory");
#endif
#endif
}

// ---------------------------------------------------------------- kernel A
__global__ void hl2_rowsum(const float* __restrict__ W, float* __restrict__ d) {
  const int i = threadIdx.x;                       // 1 block x 256 threads
  float s = 0.f;
  for (int j = 0; j < L; ++j) s += W[j * L + i];   // W symmetric: col == row sum
  d[i] = s;
}

// ------------------------------------------------------- Gram tile helper
// Computes T consecutive 16x16 Gram tiles G(I, Jb..Jb+T-1) over this block's
// K-chunk (KB rows, transposed in LDS) and contracts each with the matching
// tile of E = diag(d) - W.  Off-diagonal tiles count twice (G, E symmetric).
// I/Jb are literals at every call site -> diag handling is constant-folded.
//
// WMMA f32 16x16x4 fragment layout (32-bit A 16x4 / mirrored B 4x16):
//   lane l: M/N = l&15, VGPR g holds K = g + 2*(l>>4)
// C/D 16x16 f32: VGPR r, lane l -> M = r + 8*(l>>4), N = l&15.
template <int T>
__device__ __forceinline__ float gram_tiles(const float* lds, int I, int Jb,
                                            const float* __restrict__ W,
                                            const float* __restrict__ d,
                                            int lane) {
  const int m  = lane & 15;
  const int kh = (lane >> 4) << 1;                 // 0 or 2

  v8f acc[T];
#pragma unroll
  for (int j = 0; j < T; ++j)
#pragma unroll
    for (int r = 0; r < 8; ++r) acc[j][r] = 0.f;

  const float* aCol = lds + (I * 16 + m) * PITCH + kh;
  for (int k = 0; k < KB; k += 4) {
    v2f a;
    a[0] = aCol[k];
    a[1] = aCol[k + 1];
#pragma unroll
    for (int j = 0; j < T; ++j) {
      const float* bCol = lds + ((Jb + j) * 16 + m) * PITCH + kh;
      v2f b;
      b[0] = bCol[k];
      b[1] = bCol[k + 1];
      acc[j] = __builtin_amdgcn_wmma_f32_16x16x4_f32(
          false, a, false, b, (short)0, acc[j], false, false);
    }
  }

  // Contract finished tiles with E = diag(d) - W (W hot in L2).
  const int mrow = (lane >> 4) << 3;               // 0 or 8
  const float dval = d[I * 16 + m];                // dead unless diag tile
  float s = 0.f;
#pragma unroll
  for (int j = 0; j < T; ++j) {
    const bool diagTile = (Jb + j) == I;           // compile-time at call site
    const float* Wt = W + (I * 16 + mrow) * L + (Jb + j) * 16 + m;
    float t = 0.f;
#pragma unroll
    for (int r = 0; r < 8; ++r) {
      float c = -Wt[r * L];
      if (diagTile && ((r + mrow) == m)) c += dval;   // +d_i on G_ii
      t = fmaf(acc[j][r], c, t);
    }
    s += diagTile ? t : 2.f * t;
  }
  return s;
}

// ---------------------------------------------------------------- kernel B
__global__ void __launch_bounds__(256)
hl2_gram(const float* __restrict__ luts, const float* __restrict__ W,
         const float* __restrict__ d, float* __restrict__ partials) {
  __shared__ float lds[L * PITCH];                 // ~66.5 KB transposed chunk
  const int tid = threadIdx.x;
  const int n0  = blockIdx.x * KB;
  const float* src = luts + (size_t)n0 * L + tid;  // thread = column

#ifdef HL2_ASYNC_STAGE
  // gfx1250 async copy: memory -> LDS with per-lane LDS destination, so the
  // transpose happens in flight and data never touches VGPRs (ASYNCcnt).
  {
    float* sp = const_cast<float*>(src);
    float* tp = lds + tid * PITCH;
    for (int r = 0; r < KB; ++r) {
      __builtin_amdgcn_global_load_async_to_lds_b32(
          (gasi_t)(gasf_t)sp, (lasi_t)(lasf_t)tp, 0, 0);
      sp += L;                                     // next luts row (coalesced)
      tp += 1;                                     // next k slot in column
    }
  }
  hl2_wait_async();
#else
  {
    float* dst = lds + tid * PITCH;
    for (int r = 0; r < KB; ++r) dst[r] = src[(size_t)r * L];
  }
#endif
  __syncthreads();

  // 8 waves cover the 136 upper-triangle tiles, 17 per wave, max 8 live
  // accumulator tiles; sched fences keep the three groups from merging.
  const int w = tid >> 5, lane = tid & 31;
  float p;
  switch (w) {
    case 0: p  = gram_tiles<8>(lds,0,0,W,d,lane);   hl2_sched_fence();
            p += gram_tiles<8>(lds,0,8,W,d,lane);   hl2_sched_fence();
            p += gram_tiles<1>(lds,15,15,W,d,lane); break;
    case 1: p  = gram_tiles<8>(lds,1,1,W,d,lane);   hl2_sched_fence();
            p += gram_tiles<7>(lds,1,9,W,d,lane);   hl2_sched_fence();
            p += gram_tiles<2>(lds,14,14,W,d,lane); break;
    case 2: p  = gram_tiles<8>(lds,2,2,W,d,lane);   hl2_sched_fence();
            p += gram_tiles<6>(lds,2,10,W,d,lane);  hl2_sched_fence();
            p += gram_tiles<3>(lds,13,13,W,d,lane); break;
    case 3: p  = gram_tiles<8>(lds,3,3,W,d,lane);   hl2_sched_fence();
            p += gram_tiles<5>(lds,3,11,W,d,lane);  hl2_sched_fence();
            p += gram_tiles<4>(lds,12,12,W,d,lane); break;
    case 4: p  = gram_tiles<8>(lds,4,4,W,d,lane);   hl2_sched_fence();
            p += gram_tiles<4>(lds,4,12,W,d,lane);  hl2_sched_fence();
            p += gram_tiles<5>(lds,11,11,W,d,lane); break;
    case 5: p  = gram_tiles<8>(lds,5,5,W,d,lane);   hl2_sched_fence();
            p += gram_tiles<3>(lds,5,13,W,d,lane);  hl2_sched_fence();
            p += gram_tiles<6>(lds,10,10,W,d,lane); break;
    case 6: p  = gram_tiles<8>(lds,6,6,W,d,lane);   hl2_sched_fence();
            p += gram_tiles<2>(lds,6,14,W,d,lane);  hl2_sched_fence();
            p += gram_tiles<7>(lds,9,9,W,d,lane);   break;
    default: p = gram_tiles<8>(lds,7,7,W,d,lane);   hl2_sched_fence();
            p += gram_tiles<1>(lds,7,15,W,d,lane);  hl2_sched_fence();
            p += gram_tiles<8>(lds,8,8,W,d,lane);   break;
  }

  // Block reduction -> one partial per block. Reuse LDS.
  __syncthreads();
  lds[tid] = p;
  __syncthreads();
  for (int s = 128; s > 0; s >>= 1) {
    if (tid < s) lds[tid] += lds[tid + s];
    __syncthreads();
  }
  if (tid == 0) partials[blockIdx.x] = lds[0];
}

// ---------------------------------------------------------------- kernel C
__global__ void hl2_reduce(const float* __restrict__ partials,
                           float* __restrict__ out) {
  __shared__ float sm[256];
  float a = 0.f;
  for (int i = threadIdx.x; i < NBLK; i += 256) a += partials[i];
  sm[threadIdx.x] = a;
  __syncthreads();
  for (int s = 128; s > 0; s >>= 1) {
    if (threadIdx.x < s) sm[threadIdx.x] += sm[threadIdx.x + s];
    __syncthreads();
  }
  if (threadIdx.x == 0) out[0] = sm[0] * (1.0f / (float)NUM_LUTS);
}

// ------------------------------------------------------------------ launch
extern "C" void kernel_launch(void* const* d_in, const int* in_sizes, int n_in,
                              void* d_out, int out_size, void* d_ws,
                              size_t ws_size, hipStream_t stream) {
  const float* luts = (const float*)d_in[0];   // [65536*256] f32
  const float* W    = (const float*)d_in[1];   // [256*256] f32
  // d_in[2] = gamma, unused by the reference forward.
  float* out = (float*)d_out;                  // 1 f32

  float* d        = (float*)d_ws;              // 256 floats
  float* partials = d + L;                     // NBLK floats

  hl2_rowsum<<<1, 256, 0, stream>>>(W, d);
  hl2_gram<<<NBLK, 256, 0, stream>>>(luts, W, d, partials);
  hl2_reduce<<<1, 256, 0, stream>>>(partials, out);
}